// RNNDecoderIndivdual_33655363732231
// MI455X (gfx1250) — compile-verified
//
#include <hip/hip_runtime.h>
#include <math.h>

typedef float v2f __attribute__((ext_vector_type(2)));
typedef float v8f __attribute__((ext_vector_type(8)));

constexpr int Bb = 64;
constexpr int Tt = 2048;
constexpr int Hh = 256;
constexpr int HS = 260;    // padded LDS row stride (floats) to avoid bank conflicts
constexpr int KT = Hh / 4; // 64 k-tiles of K=4 per WMMA
constexpr int MT = 4;      // M-tiles per block in xp1_kernel

#define WMMA_F32(a, b, c) \
    __builtin_amdgcn_wmma_f32_16x16x4_f32(false, (a), false, (b), (short)0, (c), false, false)

// Split workgroup barrier: publish LDS writes, then signal; wait placed at the
// consumer so global loads / epilogue overlap barrier latency.
__device__ __forceinline__ void barrier_signal_lds() {
    asm volatile("s_wait_dscnt 0x0\n\ts_barrier_signal -1" ::: "memory");
}
__device__ __forceinline__ void barrier_wait() {
    asm volatile("s_barrier_wait -1" ::: "memory");
}

// ---------------------------------------------------------------------------
// Recurrence kernel: one block = 16 batch rows, 16 waves = 16 N-tiles (H=256).
// h state double-buffered in LDS; W_hh B-fragments held in registers.
// K=256 reduction split into 4 independent WMMA chains to shorten the
// per-step dependent-latency critical path (the serial bottleneck).
// ---------------------------------------------------------------------------
template <int LAYER>
__global__ __launch_bounds__(512) void rnn_rec_kernel(
    const float* __restrict__ xin,   // L0: x [B,T] ; L1: xp1 [B,T,H]
    const float* __restrict__ Whh,   // [H,H] row-major (n,k)
    const float* __restrict__ wih,   // L0: W_ih0 [H]
    const float* __restrict__ bih,   // L0 only
    const float* __restrict__ bhh,   // L0 only
    float* __restrict__ hout)        // [B,T,H]
{
    __shared__ float hbuf[2][16][HS];
    __shared__ float xt[2][16];

    const int wave = threadIdx.x >> 5;
    const int lane = threadIdx.x & 31;
    const int half = lane >> 4;
    const int l16  = lane & 15;
    const int n    = wave * 16 + l16;
    const int b0   = blockIdx.x * 16;

    // Preload B-fragments: B[k][n] = Whh[n][k]; lane holds N=n, K pair per half
    v2f bfrag[KT];
    {
        const float* wrow = Whh + (size_t)n * Hh + 2 * half;
#pragma unroll
        for (int k = 0; k < KT; ++k)
            bfrag[k] = *(const v2f*)(wrow + 4 * k);
    }

    float wih_n = 0.f, bias_n = 0.f;
    if constexpr (LAYER == 0) {
        wih_n  = wih[n];
        bias_n = bih[n] + bhh[n];
    }

    // zero initial hidden state (buffer 0) + stage x for t=0
    for (int i = threadIdx.x; i < 16 * HS; i += 512)
        (&hbuf[0][0][0])[i] = 0.f;
    if constexpr (LAYER == 0) {
        if (threadIdx.x < 16)
            xt[0][threadIdx.x] = xin[(size_t)(b0 + threadIdx.x) * Tt];
    }
    barrier_signal_lds();

    int cur = 0;
    for (int t = 0; t < Tt; ++t) {
        float xp[8];
        if constexpr (LAYER == 1) {
            // issue global xp loads before the barrier wait (overlap)
            const float* xprow =
                xin + ((size_t)(b0 + 8 * half) * Tt + t) * Hh + n;
#pragma unroll
            for (int r = 0; r < 8; ++r)
                xp[r] = xprow[(size_t)r * Tt * Hh];
        }

        barrier_wait();  // hbuf[cur] and xt[t&1] are now published

        // 4 independent accumulator chains over K (interleaved round-robin)
        v8f a0 = {}, a1 = {}, a2 = {}, a3 = {};
        const float* hrow = &hbuf[cur][l16][2 * half];
#pragma unroll
        for (int k = 0; k < KT / 4; ++k) {
            v2f f0 = *(const v2f*)(hrow + 4 * k);
            v2f f1 = *(const v2f*)(hrow + 4 * (k + 16));
            v2f f2 = *(const v2f*)(hrow + 4 * (k + 32));
            v2f f3 = *(const v2f*)(hrow + 4 * (k + 48));
            a0 = WMMA_F32(f0, bfrag[k], a0);
            a1 = WMMA_F32(f1, bfrag[k + 16], a1);
            a2 = WMMA_F32(f2, bfrag[k + 32], a2);
            a3 = WMMA_F32(f3, bfrag[k + 48], a3);
        }
        v8f c = (a0 + a1) + (a2 + a3);

        const int nxt = cur ^ 1;
#pragma unroll
        for (int r = 0; r < 8; ++r) {
            const int m = r + 8 * half;
            float pre;
            if constexpr (LAYER == 0)
                pre = c[r] + xt[t & 1][m] * wih_n + bias_n;
            else
                pre = c[r] + xp[r];
            const float hv = tanhf(pre);
            hbuf[nxt][m][n] = hv;
            hout[((size_t)(b0 + m) * Tt + t) * Hh + n] = hv;
        }
        if constexpr (LAYER == 0) {
            if (threadIdx.x < 16 && t + 1 < Tt)
                xt[(t + 1) & 1][threadIdx.x] =
                    xin[(size_t)(b0 + threadIdx.x) * Tt + t + 1];
        }
        if (t + 1 < Tt) barrier_signal_lds();
        cur = nxt;
    }
}

// ---------------------------------------------------------------------------
// xp1 = h0 @ W_ih1^T + b_ih1 + b_hh1 over flattened rows (B*T, H).
// Block = MT 16-row M-tiles (B-fragments amortized); 16 waves = 16 N-tiles.
// ---------------------------------------------------------------------------
__global__ __launch_bounds__(512) void xp1_kernel(
    const float* __restrict__ h0,    // [B*T, H]
    const float* __restrict__ Wih1,  // [H,H] (n,k)
    const float* __restrict__ bih1,
    const float* __restrict__ bhh1,
    float* __restrict__ xp1)         // [B*T, H]
{
    __shared__ float As[16][HS];

    const int wave = threadIdx.x >> 5;
    const int lane = threadIdx.x & 31;
    const int half = lane >> 4;
    const int l16  = lane & 15;
    const int n    = wave * 16 + l16;

    // B-fragments from W_ih1 (L2-resident, 256 KB), reused across MT tiles
    v2f bfrag[KT];
    {
        const float* wrow = Wih1 + (size_t)n * Hh + 2 * half;
#pragma unroll
        for (int k = 0; k < KT; ++k)
            bfrag[k] = *(const v2f*)(wrow + 4 * k);
    }
    const float bsum = bih1[n] + bhh1[n];

    for (int it = 0; it < MT; ++it) {
        const size_t r0 = ((size_t)blockIdx.x * MT + it) * 16;

        // cooperative, coalesced A-slab load: 16 rows x 256 cols
        for (int i = threadIdx.x; i < 16 * Hh; i += 512) {
            const int rr = i >> 8, cc = i & 255;
            As[rr][cc] = h0[(r0 + rr) * Hh + cc];
        }
        __syncthreads();

        v8f a0 = {}, a1 = {}, a2 = {}, a3 = {};
        const float* arow = &As[l16][2 * half];
#pragma unroll
        for (int k = 0; k < KT / 4; ++k) {
            v2f f0 = *(const v2f*)(arow + 4 * k);
            v2f f1 = *(const v2f*)(arow + 4 * (k + 16));
            v2f f2 = *(const v2f*)(arow + 4 * (k + 32));
            v2f f3 = *(const v2f*)(arow + 4 * (k + 48));
            a0 = WMMA_F32(f0, bfrag[k], a0);
            a1 = WMMA_F32(f1, bfrag[k + 16], a1);
            a2 = WMMA_F32(f2, bfrag[k + 32], a2);
            a3 = WMMA_F32(f3, bfrag[k + 48], a3);
        }
        v8f c = (a0 + a1) + (a2 + a3);

#pragma unroll
        for (int r = 0; r < 8; ++r) {
            const int m = r + 8 * half;
            xp1[(r0 + m) * Hh + n] = c[r] + bsum;
        }
        __syncthreads();  // As reused next iteration
    }
}

// ---------------------------------------------------------------------------
// y[b,t] = z[b,t] * dot(h1[b,t,:], W_lin) + b_lin
// out[b, 0..T-1] = y ; out[b, T] = max_t y
// ---------------------------------------------------------------------------
__global__ __launch_bounds__(256) void out_kernel(
    const float* __restrict__ h1,    // [B,T,H]
    const float* __restrict__ z,     // [B,T]
    const float* __restrict__ Wlin,  // [H]
    const float* __restrict__ blin,  // [1]
    float* __restrict__ out)         // [B, T+1]
{
    __shared__ float smax[8];
    const int b    = blockIdx.x;
    const int wave = threadIdx.x >> 5;
    const int lane = threadIdx.x & 31;

    float wl[8];
#pragma unroll
    for (int j = 0; j < 8; ++j) wl[j] = Wlin[lane + 32 * j];
    const float bl = blin[0];

    float vmax = -INFINITY;
    for (int t = wave; t < Tt; t += 8) {
        const float* hr = h1 + ((size_t)b * Tt + t) * Hh;
        float p = 0.f;
#pragma unroll
        for (int j = 0; j < 8; ++j) p += hr[lane + 32 * j] * wl[j];
#pragma unroll
        for (int off = 16; off > 0; off >>= 1)
            p += __shfl_xor(p, off, 32);
        const float y = z[(size_t)b * Tt + t] * p + bl;
        if (lane == 0) out[(size_t)b * (Tt + 1) + t] = y;
        vmax = fmaxf(vmax, y);
    }
    if (lane == 0) smax[wave] = vmax;
    __syncthreads();
    if (threadIdx.x == 0) {
        float m = smax[0];
#pragma unroll
        for (int w = 1; w < 8; ++w) m = fmaxf(m, smax[w]);
        out[(size_t)b * (Tt + 1) + Tt] = m;
    }
}

// ---------------------------------------------------------------------------
extern "C" void kernel_launch(void* const* d_in, const int* in_sizes, int n_in,
                              void* d_out, int out_size, void* d_ws, size_t ws_size,
                              hipStream_t stream) {
    const float* x    = (const float*)d_in[0];
    const float* z    = (const float*)d_in[1];
    const float* Wih0 = (const float*)d_in[2];
    const float* Whh0 = (const float*)d_in[3];
    const float* bih0 = (const float*)d_in[4];
    const float* bhh0 = (const float*)d_in[5];
    const float* Wih1 = (const float*)d_in[6];
    const float* Whh1 = (const float*)d_in[7];
    const float* bih1 = (const float*)d_in[8];
    const float* bhh1 = (const float*)d_in[9];
    const float* Wlin = (const float*)d_in[10];
    const float* blin = (const float*)d_in[11];

    const size_t BTH = (size_t)Bb * Tt * Hh;
    float* hbuf   = (float*)d_ws;        // h0, later reused as h1
    float* xp1buf = hbuf + BTH;

    // layer 0 recurrence -> h0
    rnn_rec_kernel<0><<<dim3(Bb / 16), dim3(512), 0, stream>>>(
        x, Whh0, Wih0, bih0, bhh0, hbuf);

    // xp1 = h0 @ W_ih1^T + biases
    xp1_kernel<<<dim3((Bb * Tt) / (16 * MT)), dim3(512), 0, stream>>>(
        hbuf, Wih1, bih1, bhh1, xp1buf);

    // layer 1 recurrence -> h1 (overwrites h0 buffer)
    rnn_rec_kernel<1><<<dim3(Bb / 16), dim3(512), 0, stream>>>(
        xp1buf, Whh1, nullptr, nullptr, nullptr, hbuf);

    // gated projection + time-max
    out_kernel<<<dim3(Bb), dim3(256), 0, stream>>>(
        hbuf, z, Wlin, blin, (float*)d_out);
}